// MVDR_Oracle_no_steering_vector_using_target_68135361183975
// MI455X (gfx1250) — compile-verified
//
#include <hip/hip_runtime.h>
#include <cstdint>

typedef __attribute__((ext_vector_type(2))) float v2f;
typedef __attribute__((ext_vector_type(8))) float v8f;

#define N_ 4
#define C_ 8
#define T_ 256
#define F_ 513
#define FBLK 32
#define NFB 17            // ceil(513/32)
#define TC 8              // t-chunk staged per iteration
#define THALF 128         // each workgroup covers half of T
// reference adds load=(0.001/sqrt(2))*(1+1j) to the T-averaged cov; we keep raw
// sums (no /T), so scale the loading by T.
#define TLOAD (7.0710678118654752e-4f * 256.0f)

// --------------------------------------------------------------------------
// Kernel 1a: coalesced async-LDS-staged covariance accumulation.
// Grid = N * 17 * 2 (t-halves). Workgroup = 1024 threads (32 waves), one n,
// 32 consecutive f, 128 timesteps. Each wave w owns f = f0 + w and
// accumulates partial D = M M^T (16x16 f32) for target and noise via
// V_WMMA_F32_16X16X4_F32 (M rows = [Re c, Im c], memory row = n*16 + m).
// Staging uses GLOBAL_LOAD_ASYNC_TO_LDS_B32 (ASYNCcnt) -- no VGPR round-trip.
// --------------------------------------------------------------------------
__global__ __launch_bounds__(1024) void mvdr_cov(
    const float* __restrict__ target,
    const float* __restrict__ noise,
    float* __restrict__ dsum)
{
    const int nb   = blockIdx.x;          // 0 .. N*NFB*2-1
    const int half = nb & 1;
    const int rest = nb >> 1;
    const int n    = rest / NFB;
    const int f0   = (rest % NFB) * FBLK;
    const int tbase = half * THALF;
    const int tid  = threadIdx.x;
    const int wave = tid >> 5;            // f lane-group: f = f0 + wave
    const int lane = tid & 31;

    // staging tiles: [t][row][f padded to 33] -> odd stride = bank-conflict-free
    __shared__ float xt[TC * 16 * 33];
    __shared__ float vt[TC * 16 * 33];

    // staging mapping: f_l fastest for coalescing
    const int f_l = tid & 31;
    const int q   = tid >> 5;             // 0..31 over (t,row) pairs
    const bool fok = (f0 + f_l) < F_;

    // WMMA A-operand mapping (16x4 f32): lane holds M = lane%16,
    // VGPR v holds K = 2*(lane/16)+v.
    const int m = lane & 15;
    const int g = lane >> 4;
    const bool fcomp = (f0 + wave) < F_;  // wave-uniform

    const uint32_t xbase = (uint32_t)(uintptr_t)xt;
    const uint32_t vbase = (uint32_t)(uintptr_t)vt;

    v8f accx = {};
    v8f accv = {};

    for (int tc = 0; tc < THALF; tc += TC) {
        const int t0 = tbase + tc;
        // ---- cooperative coalesced async staging: 2 * 16*8*32 floats ----
#pragma unroll
        for (int i = 0; i < 4; ++i) {
            const int pair = q + 32 * i;          // 0..127
            const int t    = pair >> 4;           // 0..7
            const int row  = pair & 15;           // 0..15
            const long ga  = ((long)(n * 16 + row) * T_ + (t0 + t)) * F_
                             + f0 + f_l;
            const uint32_t ldso = (uint32_t)(((t * 16 + row) * 33 + f_l) * 4);
            if (fok) {
                asm volatile("global_load_async_to_lds_b32 %0, %1, off"
                             :: "v"(xbase + ldso), "v"(target + ga) : "memory");
                asm volatile("global_load_async_to_lds_b32 %0, %1, off"
                             :: "v"(vbase + ldso), "v"(noise + ga) : "memory");
                if (tc + TC < THALF) {            // prefetch next chunk into L2
                    __builtin_prefetch(&target[ga + (long)TC * F_], 0, 3);
                    __builtin_prefetch(&noise[ga + (long)TC * F_], 0, 3);
                }
            }
        }
        asm volatile("s_wait_asynccnt 0x0" ::: "memory");
        __syncthreads();

        // ---- per-wave WMMA over this chunk (2 k-steps of K=4) ----
        if (fcomp) {
#pragma unroll
            for (int tb = 0; tb < TC; tb += 4) {
                const int a0 = ((tb + 2 * g) * 16 + m) * 33 + wave;
                v2f a, b;
                a[0] = xt[a0];  a[1] = xt[a0 + 16 * 33];
                b[0] = vt[a0];  b[1] = vt[a0 + 16 * 33];
                accx = __builtin_amdgcn_wmma_f32_16x16x4_f32(
                    false, a, false, a, (short)0, accx, false, false);
                accv = __builtin_amdgcn_wmma_f32_16x16x4_f32(
                    false, b, false, b, (short)0, accv, false, false);
            }
        }
        __syncthreads();
    }

    // D layout: lane holds col N = lane%16, VGPR r holds row M = r + 8*(lane/16)
    if (fcomp) {
        float* dbase = dsum + ((long)half * N_ * F_ + n * F_ + f0 + wave) * 512;
#pragma unroll
        for (int r = 0; r < 8; ++r) {
            dbase[(r + 8 * g) * 16 + m]       = accx[r];
            dbase[256 + (r + 8 * g) * 16 + m] = accv[r];
        }
    }
}

// --------------------------------------------------------------------------
// Kernel 1b: one wave per (n,f). Sum the two t-half partials, assemble complex
// Phi_x, Phi_v (+loading), lane-parallel Gauss-Jordan on [Phi_v | Phi_x] -> G,
// W = G[:,ref]/trace(G).
// --------------------------------------------------------------------------
__global__ __launch_bounds__(32) void mvdr_solve(
    const float* __restrict__ dsum,
    const int* __restrict__ refp,
    float* __restrict__ w_out)
{
    const int lane = threadIdx.x;
    const int nf = blockIdx.x;
    const int n  = nf / F_;
    const int f  = nf % F_;

    const long HS = (long)N_ * F_ * 512;
    const float* D0 = dsum + (long)nf * 512;       // half 0: [Dx(256) Dv(256)]
    const float* D1 = D0 + HS;                     // half 1

    __shared__ float Are[8][16];
    __shared__ float Aim[8][16];
    __shared__ float ltr[2];

    // Phi_re[c][d] = D[c][d] + D[c+8][d+8]; Phi_im[c][d] = D[c+8][d] - D[c][d+8]
    for (int p = lane; p < 64; p += 32) {
        const int c = p >> 3, d = p & 7;
        const int i00 = c * 16 + d,        i11 = (c + 8) * 16 + d + 8;
        const int i10 = (c + 8) * 16 + d,  i01 = c * 16 + d + 8;
        float xr = (D0[i00] + D1[i00]) + (D0[i11] + D1[i11]);
        float xi = (D0[i10] + D1[i10]) - (D0[i01] + D1[i01]);
        float vr = (D0[256 + i00] + D1[256 + i00]) + (D0[256 + i11] + D1[256 + i11]);
        float vi = (D0[256 + i10] + D1[256 + i10]) - (D0[256 + i01] + D1[256 + i01]);
        if (c == d) { vr += TLOAD; vi += TLOAD; }
        Are[c][d] = vr;      Aim[c][d] = vi;
        Are[c][d + 8] = xr;  Aim[c][d + 8] = xi;
    }
    __syncthreads();

    // Gauss-Jordan (no pivoting: Phi_v Hermitian PD + diagonal loading)
    for (int k = 0; k < 8; ++k) {
        float pr = Are[k][k], pi = Aim[k][k];
        float inv = 1.0f / (pr * pr + pi * pi);
        float ipr = pr * inv, ipi = -pi * inv;
        __syncthreads();
        if (lane < 16) {
            float ar = Are[k][lane], ai = Aim[k][lane];
            Are[k][lane] = ar * ipr - ai * ipi;
            Aim[k][lane] = ar * ipi + ai * ipr;
        }
        __syncthreads();
        float fr[8], fi[8];
#pragma unroll
        for (int i = 0; i < 8; ++i) { fr[i] = Are[i][k]; fi[i] = Aim[i][k]; }
        __syncthreads();
        for (int p = lane; p < 128; p += 32) {
            const int i = p >> 4, j = p & 15;
            if (i != k) {
                float br = Are[k][j], bi = Aim[k][j];
                Are[i][j] -= fr[i] * br - fi[i] * bi;
                Aim[i][j] -= fr[i] * bi + fi[i] * br;
            }
        }
        __syncthreads();
    }

    if (lane == 0) {
        float tr = 0.f, ti = 0.f;
#pragma unroll
        for (int c = 0; c < 8; ++c) { tr += Are[c][8 + c]; ti += Aim[c][8 + c]; }
        float inv = 1.0f / (tr * tr + ti * ti);
        ltr[0] = tr * inv;  ltr[1] = -ti * inv;
    }
    __syncthreads();
    if (lane < 8) {
        const int ref = refp[0];
        float lr = ltr[0], li = ltr[1];
        float gr = Are[lane][8 + ref], gi = Aim[lane][8 + ref];
        long o = (((long)n * F_ + f) * 8 + lane) * 2;
        w_out[o + 0] = gr * lr - gi * li;
        w_out[o + 1] = gr * li + gi * lr;
    }
}

// --------------------------------------------------------------------------
// Kernel 2: X_bf[n,t,f] = sum_c conj(W[n,f,c]) * y[n,c,t,f]  (HBM-bound stream)
// --------------------------------------------------------------------------
__global__ __launch_bounds__(256) void mvdr_apply(
    const float* __restrict__ mixture,
    const float* __restrict__ w,
    float* __restrict__ out)
{
    const long total = (long)N_ * T_ * F_;
    long idx = (long)blockIdx.x * blockDim.x + threadIdx.x;
    if (idx >= total) return;
    const int f = (int)(idx % F_);
    const int t = (int)((idx / F_) % T_);
    const int n = (int)(idx / ((long)T_ * F_));

    const float4* Wp = (const float4*)(w + ((long)n * F_ + f) * 16);
    float4 w0 = Wp[0], w1 = Wp[1], w2 = Wp[2], w3 = Wp[3];
    const float wr[8] = {w0.x, w0.z, w1.x, w1.z, w2.x, w2.z, w3.x, w3.z};
    const float wi[8] = {w0.y, w0.w, w1.y, w1.w, w2.y, w2.w, w3.y, w3.w};

    const long tf = (long)t * F_ + f;
    const long plane = (long)T_ * F_;
    const float* yre = mixture + (long)(n * 2 + 0) * C_ * plane + tf;
    const float* yim = mixture + (long)(n * 2 + 1) * C_ * plane + tf;

    float accr = 0.f, acci = 0.f;
#pragma unroll
    for (int c = 0; c < 8; ++c) {
        float yr = yre[(long)c * plane];
        float yi = yim[(long)c * plane];
        accr += wr[c] * yr + wi[c] * yi;   // conj(W)*y
        acci += wr[c] * yi - wi[c] * yr;
    }
    out[(long)(n * 2 + 0) * plane + tf] = accr;
    out[(long)(n * 2 + 1) * plane + tf] = acci;
}

extern "C" void kernel_launch(void* const* d_in, const int* in_sizes, int n_in,
                              void* d_out, int out_size, void* d_ws, size_t ws_size,
                              hipStream_t stream) {
    const float* mixture = (const float*)d_in[0];
    const float* target  = (const float*)d_in[1];
    const float* noise   = (const float*)d_in[2];
    const int*   refp    = (const int*)d_in[3];
    float* out  = (float*)d_out;

    float* dsum = (float*)d_ws;                         // 2 x (N*F,2,16,16)
    float* wws  = dsum + 2L * N_ * F_ * 512;            // (N, F, C, 2) weights

    mvdr_cov<<<N_ * NFB * 2, 1024, 0, stream>>>(target, noise, dsum);
    mvdr_solve<<<N_ * F_, 32, 0, stream>>>(dsum, refp, wws);

    const long total = (long)N_ * T_ * F_;
    const int blocks = (int)((total + 255) / 256);
    mvdr_apply<<<blocks, 256, 0, stream>>>(mixture, wws, out);
}